// GatingNetwork_86268713107988
// MI455X (gfx1250) — compile-verified
//
#include <hip/hip_runtime.h>
#include <hip/hip_bf16.h>

typedef __attribute__((ext_vector_type(2))) float v2f;
typedef __attribute__((ext_vector_type(8))) float v8f;

#define HIDDEN 4096
#define NEXP 64
#define NTOK 16384
#define TOK_PER_BLOCK 128   // 8 waves * 16 tokens
#define LDS_STRIDE 65       // 64 experts + 1 pad -> conflict-free LDS

__device__ __forceinline__ v8f splat8(float v) {
    v8f r;
#pragma unroll
    for (int i = 0; i < 8; ++i) r[i] = v;
    return r;
}

__global__ __launch_bounds__(256) void gating_topk_kernel(
    const float* __restrict__ x,   // [NTOK, HIDDEN]
    const float* __restrict__ W,   // [NEXP, HIDDEN]
    const float* __restrict__ b,   // [NEXP]
    float* __restrict__ out_vals,  // [NTOK, 2]
    int*   __restrict__ out_idx)   // [NTOK, 2]
{
    __shared__ float logits_lds[TOK_PER_BLOCK * LDS_STRIDE];

    const int tid  = threadIdx.x;
    const int wave = tid >> 5;
    const int lane = tid & 31;
    const int half = lane >> 4;   // 0: lanes 0-15, 1: lanes 16-31
    const int lmod = lane & 15;

    const int blockTok = blockIdx.x * TOK_PER_BLOCK;
    const int tokBase  = blockTok + wave * 16;

    // ---- Phase 1: 16x64 logits tile via V_WMMA_F32_16X16X4_F32 ----
    // A (16x4 f32): lane holds row M=lane%16; VGPR v <-> K = v + 2*(lane/16)
    // B (4x16 f32): lane holds col N=lane%16; VGPR v <-> K = v + 2*(lane/16)
    const float2* xk = reinterpret_cast<const float2*>(
        x + (size_t)(tokBase + lmod) * HIDDEN + 2 * half);
    const float2* wk0 = reinterpret_cast<const float2*>(
        W + (size_t)( 0 + lmod) * HIDDEN + 2 * half);
    const float2* wk1 = reinterpret_cast<const float2*>(
        W + (size_t)(16 + lmod) * HIDDEN + 2 * half);
    const float2* wk2 = reinterpret_cast<const float2*>(
        W + (size_t)(32 + lmod) * HIDDEN + 2 * half);
    const float2* wk3 = reinterpret_cast<const float2*>(
        W + (size_t)(48 + lmod) * HIDDEN + 2 * half);

    // Seed accumulators with the bias: every C/D VGPR of a lane maps to
    // expert column N = lmod (+16*g), constant across the 8 row-registers.
    v8f acc0 = splat8(b[ 0 + lmod]);
    v8f acc1 = splat8(b[16 + lmod]);
    v8f acc2 = splat8(b[32 + lmod]);
    v8f acc3 = splat8(b[48 + lmod]);

#pragma unroll 8
    for (int k2 = 0; k2 < HIDDEN / 2; k2 += 2) {   // K advances by 4 per iter
        float2 a  = xk[k2];
        float2 b0 = wk0[k2];
        float2 b1 = wk1[k2];
        float2 b2 = wk2[k2];
        float2 b3 = wk3[k2];
        v2f af  = {a.x,  a.y};
        v2f bf0 = {b0.x, b0.y};
        v2f bf1 = {b1.x, b1.y};
        v2f bf2 = {b2.x, b2.y};
        v2f bf3 = {b3.x, b3.y};
        acc0 = __builtin_amdgcn_wmma_f32_16x16x4_f32(false, af, false, bf0,
                                                     (short)0, acc0, false, false);
        acc1 = __builtin_amdgcn_wmma_f32_16x16x4_f32(false, af, false, bf1,
                                                     (short)0, acc1, false, false);
        acc2 = __builtin_amdgcn_wmma_f32_16x16x4_f32(false, af, false, bf2,
                                                     (short)0, acc2, false, false);
        acc3 = __builtin_amdgcn_wmma_f32_16x16x4_f32(false, af, false, bf3,
                                                     (short)0, acc3, false, false);
    }

    // C/D layout: VGPR r -> token row M = r + 8*half; column N = lmod
#pragma unroll
    for (int r = 0; r < 8; ++r) {
        const int trel = wave * 16 + r + 8 * half;
        float* row = &logits_lds[trel * LDS_STRIDE];
        row[ 0 + lmod] = acc0[r];
        row[16 + lmod] = acc1[r];
        row[32 + lmod] = acc2[r];
        row[48 + lmod] = acc3[r];
    }
    __syncthreads();

    // ---- Phase 2: softmax + top-2 per token (threads 0..127) ----
    if (tid < TOK_PER_BLOCK) {
        const float* row = &logits_lds[tid * LDS_STRIDE];
        float m1 = -3.402823466e+38f; int i1 = 0;
        float m2 = -3.402823466e+38f; int i2 = 0;
#pragma unroll 8
        for (int e = 0; e < NEXP; ++e) {
            float v = row[e];
            if (v > m1) { m2 = m1; i2 = i1; m1 = v; i1 = e; }
            else if (v > m2) { m2 = v; i2 = e; }
        }
        float s = 0.0f;
#pragma unroll 8
        for (int e = 0; e < NEXP; ++e) s += __expf(row[e] - m1);
        const float inv_s = 1.0f / s;

        const int tok = blockTok + tid;
        out_vals[tok * 2 + 0] = inv_s;                  // exp(m1-m1)/s
        out_vals[tok * 2 + 1] = __expf(m2 - m1) * inv_s;
        out_idx [tok * 2 + 0] = i1;
        out_idx [tok * 2 + 1] = i2;
    }
}

extern "C" void kernel_launch(void* const* d_in, const int* in_sizes, int n_in,
                              void* d_out, int out_size, void* d_ws, size_t ws_size,
                              hipStream_t stream) {
    const float* x = (const float*)d_in[0];
    const float* W = (const float*)d_in[1];
    const float* b = (const float*)d_in[2];

    float* out_vals = (float*)d_out;                          // [NTOK,2] f32
    int*   out_idx  = (int*)((float*)d_out + (size_t)NTOK*2); // [NTOK,2] i32

    dim3 grid(NTOK / TOK_PER_BLOCK);
    dim3 block(256);
    gating_topk_kernel<<<grid, block, 0, stream>>>(x, W, b, out_vals, out_idx);
}